// BlockAttentionResidual_38517266710763
// MI455X (gfx1250) — compile-verified
//
#include <hip/hip_runtime.h>
#include <math.h>
#include <stdint.h>

typedef __attribute__((ext_vector_type(2))) float v2f;
typedef __attribute__((ext_vector_type(8))) float v8f;

#define N_DEPTH 8
#define DIM     2048
#define THREADS 256
#define WAVES   (THREADS / 32)
#define EPS_F   1e-6f

// Dynamic LDS layout (floats):
// [0, N_DEPTH*DIM)            : staged V rows for this token (8 x 2048 f32 = 64 KiB)
// [N_DEPTH*DIM, +WAVES*16)    : per-wave reduction partials (8 waves x {8 dot, 8 ss})
// [.., +16)                   : reduced totals (8 dot, 8 ss)
#define LDS_FLOATS (N_DEPTH * DIM + WAVES * 16 + 16)

__global__ __launch_bounds__(THREADS) void
block_attn_residual_kernel(const float* __restrict__ V,
                           const float* __restrict__ w,
                           const float* __restrict__ q,
                           float* __restrict__ out,
                           int tokens)
{
    extern __shared__ float smem[];
    float* ldsV   = smem;                         // 8 rows x 2048
    float* ldsRed = smem + N_DEPTH * DIM;         // [wave][16]
    float* ldsTot = ldsRed + WAVES * 16;          // [16]

    const int tid   = threadIdx.x;
    const int lane  = tid & 31;
    const int wave  = tid >> 5;
    const int token = blockIdx.x;

    // ---------------- Async stage: wave n DMAs row n (8 KiB) into LDS -------------
    {
        const int n = wave;  // 8 waves <-> 8 depth entries
        // byte offset of row (n, token) inside V; max 512 MiB fits in i32
        unsigned long long rowElem =
            ((unsigned long long)n * (unsigned long long)tokens +
             (unsigned long long)token) * (unsigned long long)DIM;
        unsigned int voff  = (unsigned int)(rowElem * 4ull) + (unsigned int)(lane * 16);
        unsigned int laddr = (unsigned int)(unsigned long long)(uintptr_t)(void*)(ldsV + n * DIM)
                             + (unsigned int)(lane * 16);
        // 16 x (32 lanes x 16B) = 8 KiB per wave, ASYNCcnt-tracked, GVS addressing
        #pragma unroll
        for (int i = 0; i < 16; ++i) {
            asm volatile("global_load_async_to_lds_b128 %0, %1, %2"
                         :: "v"(laddr), "v"(voff), "s"(V)
                         : "memory");
            laddr += 512;
            voff  += 512;
        }
    }

    // Overlap: load this thread's slice of (w * q) while the async engine streams V.
    float4 wqa, wqb;
    {
        const float4* w4 = (const float4*)w;
        const float4* q4 = (const float4*)q;
        float4 wv = w4[tid];       float4 qv = q4[tid];
        wqa.x = wv.x * qv.x; wqa.y = wv.y * qv.y; wqa.z = wv.z * qv.z; wqa.w = wv.w * qv.w;
        wv = w4[tid + THREADS];    qv = q4[tid + THREADS];
        wqb.x = wv.x * qv.x; wqb.y = wv.y * qv.y; wqb.z = wv.z * qv.z; wqb.w = wv.w * qv.w;
    }

    asm volatile("s_wait_asynccnt 0x0" ::: "memory");
    __syncthreads();

    // ---------------- Per-thread partial dot / sumsq over its d-slice --------------
    float4 va[N_DEPTH], vb[N_DEPTH];
    float pdot[N_DEPTH], pss[N_DEPTH];
    const float4* lv4 = (const float4*)ldsV;
    #pragma unroll
    for (int n = 0; n < N_DEPTH; ++n) {
        float4 a = lv4[n * (DIM / 4) + tid];
        float4 b = lv4[n * (DIM / 4) + THREADS + tid];
        va[n] = a; vb[n] = b;
        pdot[n] = a.x * wqa.x + a.y * wqa.y + a.z * wqa.z + a.w * wqa.w
                + b.x * wqb.x + b.y * wqb.y + b.z * wqb.z + b.w * wqb.w;
        pss[n]  = a.x * a.x + a.y * a.y + a.z * a.z + a.w * a.w
                + b.x * b.x + b.y * b.y + b.z * b.z + b.w * b.w;
    }

    // ---------------- Intra-wave butterfly reduction (wave32) ---------------------
    #pragma unroll
    for (int n = 0; n < N_DEPTH; ++n) {
        float d = pdot[n], s = pss[n];
        #pragma unroll
        for (int off = 16; off > 0; off >>= 1) {
            d += __shfl_xor(d, off, 32);
            s += __shfl_xor(s, off, 32);
        }
        pdot[n] = d; pss[n] = s;
    }
    if (lane == 0) {
        #pragma unroll
        for (int n = 0; n < N_DEPTH; ++n) {
            ldsRed[wave * 16 + n]     = pdot[n];
            ldsRed[wave * 16 + 8 + n] = pss[n];
        }
    }
    __syncthreads();

    // ---------------- Cross-wave reduction via V_WMMA_F32_16X16X4_F32 -------------
    // D = 1(16x4) x P(4x16) + C  => every row of D holds the column sums of P.
    // Two accumulating WMMAs cover all 8 wave-partial rows.
    if (wave == 0) {
        v2f A; A[0] = 1.0f; A[1] = 1.0f;            // all-ones A matrix
        const int col = lane & 15;
        const int hi  = lane >> 4;                  // lanes 16-31 carry K=2/3 halves
        v2f B0, B1;
        B0[0] = ldsRed[(hi ? 2 : 0) * 16 + col];    // K=0 / K=2
        B0[1] = ldsRed[(hi ? 3 : 1) * 16 + col];    // K=1 / K=3
        B1[0] = ldsRed[(hi ? 6 : 4) * 16 + col];    // K=4 / K=6
        B1[1] = ldsRed[(hi ? 7 : 5) * 16 + col];    // K=5 / K=7
        v8f C = {};
        C = __builtin_amdgcn_wmma_f32_16x16x4_f32(false, A, false, B0, (short)0, C, false, false);
        C = __builtin_amdgcn_wmma_f32_16x16x4_f32(false, A, false, B1, (short)0, C, false, false);
        if (lane < 16) ldsTot[lane] = C[0];         // row M=0: totals per quantity
    }
    __syncthreads();

    // ---------------- Logits, softmax over n, weighted combine ---------------------
    float lg[N_DEPTH];
    float mx = -INFINITY;
    #pragma unroll
    for (int n = 0; n < N_DEPTH; ++n) {
        float dot = ldsTot[n];
        float ss  = ldsTot[8 + n];
        float r   = rsqrtf(ss * (1.0f / (float)DIM) + EPS_F);
        lg[n] = dot * r;                            // sum(V * w * q) / rms
        mx = fmaxf(mx, lg[n]);
    }
    float se = 0.0f;
    #pragma unroll
    for (int n = 0; n < N_DEPTH; ++n) {
        lg[n] = expf(lg[n] - mx);
        se += lg[n];
    }
    const float inv = 1.0f / se;

    float4 oa = {0.f, 0.f, 0.f, 0.f}, ob = {0.f, 0.f, 0.f, 0.f};
    #pragma unroll
    for (int n = 0; n < N_DEPTH; ++n) {
        const float wn = lg[n] * inv;
        oa.x += wn * va[n].x; oa.y += wn * va[n].y; oa.z += wn * va[n].z; oa.w += wn * va[n].w;
        ob.x += wn * vb[n].x; ob.y += wn * vb[n].y; ob.z += wn * vb[n].z; ob.w += wn * vb[n].w;
    }

    float4* o4 = (float4*)(out + (size_t)token * DIM);
    o4[tid] = oa;
    o4[tid + THREADS] = ob;
}

extern "C" void kernel_launch(void* const* d_in, const int* in_sizes, int n_in,
                              void* d_out, int out_size, void* d_ws, size_t ws_size,
                              hipStream_t stream) {
    (void)n_in; (void)d_ws; (void)ws_size; (void)in_sizes;
    const float* V = (const float*)d_in[0];   // (n, B, T, D) f32
    const float* w = (const float*)d_in[1];   // (D,) f32
    const float* q = (const float*)d_in[2];   // (D,) f32
    float* out = (float*)d_out;               // (B, T, D) f32

    const int tokens = out_size / DIM;        // B*T = 8192
    const size_t lds_bytes = LDS_FLOATS * sizeof(float);  // ~66 KiB dynamic LDS

    hipLaunchKernelGGL(block_attn_residual_kernel,
                       dim3(tokens), dim3(THREADS), lds_bytes, stream,
                       V, w, q, out, tokens);
}